// RBFLayer_6777458393496
// MI455X (gfx1250) — compile-verified
//
#include <hip/hip_runtime.h>

typedef __attribute__((ext_vector_type(2))) float v2f;
typedef __attribute__((ext_vector_type(4))) float v4f;
typedef __attribute__((ext_vector_type(8))) float v8f;

#define LDSTRIDE 68   // 64 + 4 pad: row stride mod 64 banks = 4 -> conflict-free frags

__global__ __launch_bounds__(256) void rbf_wmma_f32_kernel(
    const float* __restrict__ x,   // (B, 64)
    const float* __restrict__ s,   // (O, 64)
    const float* __restrict__ e,   // (O,)
    float* __restrict__ out,       // (B, O)
    int O)
{
    __shared__ float xs[128 * LDSTRIDE];   // x tile: 128 rows x 64 (padded)
    __shared__ float ss[64 * LDSTRIDE];    // s tile:  64 rows x 64 (padded)
    __shared__ float x2s[128];             // ||x_b||^2 per tile row
    __shared__ float s2s[64];              // ||s_o||^2 per tile col
    __shared__ float es[64];               // e per tile col

    const int tid = threadIdx.x;
    const int b0  = blockIdx.x * 128;      // batch tile origin
    const int o0  = blockIdx.y * 64;       // output tile origin

    // ---- Phase 1: cooperative global -> LDS tile load (float4) ----
    {
        const v4f* xg = (const v4f*)(x + (size_t)b0 * 64);
        #pragma unroll
        for (int i = 0; i < 8; ++i) {
            int idx = tid + i * 256;           // 0..2047 float4s in 128x64 tile
            int row = idx >> 4;                // 16 float4 per row
            int c4  = idx & 15;
            v4f v = xg[idx];
            *(v4f*)&xs[row * LDSTRIDE + c4 * 4] = v;
        }
        const v4f* sg = (const v4f*)(s + (size_t)o0 * 64);
        #pragma unroll
        for (int i = 0; i < 4; ++i) {
            int idx = tid + i * 256;           // 0..1023 float4s in 64x64 tile
            int row = idx >> 4;
            int c4  = idx & 15;
            v4f v = sg[idx];
            *(v4f*)&ss[row * LDSTRIDE + c4 * 4] = v;
        }
    }
    __syncthreads();

    // ---- Phase 2: row norms + betas into LDS ----
    if (tid < 128) {
        const float* r = &xs[tid * LDSTRIDE];
        float a = 0.f;
        #pragma unroll
        for (int k = 0; k < 64; ++k) a += r[k] * r[k];
        x2s[tid] = a;
    } else if (tid < 192) {
        int r0 = tid - 128;
        const float* r = &ss[r0 * LDSTRIDE];
        float a = 0.f;
        #pragma unroll
        for (int k = 0; k < 64; ++k) a += r[k] * r[k];
        s2s[r0] = a;
    } else {
        int c = tid - 192;
        es[c] = e[o0 + c];
    }
    __syncthreads();

    // ---- Phase 3: cross term via V_WMMA_F32_16X16X4_F32 ----
    const int lane = tid & 31;
    const int wave = tid >> 5;
    const int mw = (wave >> 1) * 32;    // 4 waves along M -> 128 rows
    const int nw = (wave & 1) * 32;     // 2 waves along N -> 64 cols
    const int li = lane & 15;
    const int kh = (lane >> 4) * 2;     // lanes 16-31 hold K+2 per ISA layout

    // A frag (16x4 f32): lane li = row M, VGPR v holds K = k0 + kh + v  (consecutive -> b64)
    const float* ax0 = &xs[(mw      + li) * LDSTRIDE + kh];
    const float* ax1 = &xs[(mw + 16 + li) * LDSTRIDE + kh];
    // B frag (4x16 f32): B[k][n] = s[n][k]; identical per-lane addressing into s tile
    const float* bx0 = &ss[(nw      + li) * LDSTRIDE + kh];
    const float* bx1 = &ss[(nw + 16 + li) * LDSTRIDE + kh];

    v8f acc00 = {}, acc01 = {}, acc10 = {}, acc11 = {};

    #pragma unroll
    for (int k0 = 0; k0 < 64; k0 += 4) {
        v2f a0 = *(const v2f*)(ax0 + k0);
        v2f a1 = *(const v2f*)(ax1 + k0);
        v2f b0 = *(const v2f*)(bx0 + k0);
        v2f b1 = *(const v2f*)(bx1 + k0);
        acc00 = __builtin_amdgcn_wmma_f32_16x16x4_f32(false, a0, false, b0,
                                                      (short)0, acc00, false, false);
        acc01 = __builtin_amdgcn_wmma_f32_16x16x4_f32(false, a0, false, b1,
                                                      (short)0, acc01, false, false);
        acc10 = __builtin_amdgcn_wmma_f32_16x16x4_f32(false, a1, false, b0,
                                                      (short)0, acc10, false, false);
        acc11 = __builtin_amdgcn_wmma_f32_16x16x4_f32(false, a1, false, b1,
                                                      (short)0, acc11, false, false);
    }

    // ---- Phase 4: epilogue  d2 = x2 + s2 - 2*cross ; out = exp(-e*d2) ----
    // C/D layout: VGPR i -> row M = i + 8*(lane>=16), col N = lane&15
    // Output is written once and never re-read: use non-temporal stores so the
    // 256 MB stream does not evict the hot x/s tiles from L2 (inputs are
    // re-read by every block along the other grid dim; they fit in 192 MB L2).
    const int rowadd = (lane >> 4) * 8;
    v8f accs[2][2];
    accs[0][0] = acc00; accs[0][1] = acc01; accs[1][0] = acc10; accs[1][1] = acc11;

    #pragma unroll
    for (int ms = 0; ms < 2; ++ms) {
        #pragma unroll
        for (int ns = 0; ns < 2; ++ns) {
            int nl = nw + ns * 16 + li;
            float s2v = s2s[nl];
            float ev  = es[nl];
            size_t ocol = (size_t)(o0 + nl);
            #pragma unroll
            for (int i = 0; i < 8; ++i) {
                int ml = mw + ms * 16 + rowadd + i;
                float d2 = x2s[ml] + s2v - 2.0f * accs[ms][ns][i];
                float r  = __expf(-ev * d2);
                __builtin_nontemporal_store(r, &out[(size_t)(b0 + ml) * (size_t)O + ocol]);
            }
        }
    }
}

extern "C" void kernel_launch(void* const* d_in, const int* in_sizes, int n_in,
                              void* d_out, int out_size, void* d_ws, size_t ws_size,
                              hipStream_t stream) {
    (void)n_in; (void)out_size; (void)d_ws; (void)ws_size;
    const float* x = (const float*)d_in[0];   // (B, 64) fp32
    const float* s = (const float*)d_in[1];   // (O, 64) fp32
    const float* e = (const float*)d_in[2];   // (O,)    fp32
    float* out = (float*)d_out;               // (B, O)  fp32

    const int D = 64;
    const int B = in_sizes[0] / D;            // 16384
    const int O = in_sizes[2];                // 4096

    dim3 grid(B / 128, O / 64);
    dim3 block(256);
    rbf_wmma_f32_kernel<<<grid, block, 0, stream>>>(x, s, e, out, O);
}